// NnBoard768Cuda_36498632081979
// MI455X (gfx1250) — compile-verified
//
#include <hip/hip_runtime.h>
#include <hip/hip_bf16.h>
#include <math.h>

// Problem constants (from reference)
#define NN_B      16384
#define NN_MAXF   32
#define NN_NFEAT  768
#define NN_FTOUT  1024   // per-slice width; hidden = 2*NN_FTOUT

// ---- gfx1250 async global->LDS (guarded so compilation never breaks) ----
#if defined(__has_builtin)
#if __has_builtin(__builtin_amdgcn_global_load_async_to_lds_b128)
#define NN_HAVE_ASYNC_LDS 1
#endif
#if __has_builtin(__builtin_amdgcn_s_wait_asynccnt)
#define NN_HAVE_WAIT_ASYNC 1
#endif
#endif

// Builtin signature (from hipcc diagnostic): arg0 is `int4 addrspace(1)*`,
// arg1 is the LDS destination, then imm offset + imm cpol.
typedef int nn_v4i __attribute__((ext_vector_type(4)));
typedef __attribute__((address_space(1))) nn_v4i* nn_gptr;   // global int4*
typedef __attribute__((address_space(3))) nn_v4i* nn_lptr;   // LDS int4*

__device__ __forceinline__ float nn_clamp01(float x) {
    return fminf(fmaxf(x, 0.0f), 1.0f);   // expect v_med3_f32
}

__global__ __launch_bounds__(256) void nnue_fused_kernel(
    const float* __restrict__ values,      // [B, 32]
    const int*   __restrict__ stm_idx,     // [B, 32]
    const int*   __restrict__ nstm_idx,    // [B, 32]
    const float* __restrict__ ft_w,        // [768, 1024]
    const float* __restrict__ ft_b,        // [1024]
    const float* __restrict__ out_w,       // [2048]
    const float* __restrict__ out_b,       // [1]
    float*       __restrict__ out)         // [B]
{
    __shared__ __align__(16) float s_outw[2 * NN_FTOUT];  // 8 KB staged 2nd-layer weights
    __shared__ float s_part[8];                           // 8 wave32 partials

    const int b   = blockIdx.x;        // one block per sample
    const int tid = threadIdx.x;       // 256 threads
    const int col = tid << 2;          // 4 floats per thread -> 1024 columns

    // ---- Stage out_w into LDS asynchronously; overlapped with the gather loop ----
#ifdef NN_HAVE_ASYNC_LDS
    __builtin_amdgcn_global_load_async_to_lds_b128(
        (nn_gptr)(out_w + col),            (nn_lptr)(s_outw + col),            0, 0);
    __builtin_amdgcn_global_load_async_to_lds_b128(
        (nn_gptr)(out_w + NN_FTOUT + col), (nn_lptr)(s_outw + NN_FTOUT + col), 0, 0);
#else
    *(float4*)(s_outw + col)            = *(const float4*)(out_w + col);
    *(float4*)(s_outw + NN_FTOUT + col) = *(const float4*)(out_w + NN_FTOUT + col);
#endif

    // ---- Feature-transformer gather-accumulate ----
    // Per-sample indices/weights are uniform across the block -> scalar (s_load) path.
    const float* vrow = values   + (size_t)b * NN_MAXF;
    const int*   srow = stm_idx  + (size_t)b * NN_MAXF;
    const int*   nrow = nstm_idx + (size_t)b * NN_MAXF;

    float4 acc_s = *(const float4*)(ft_b + col);
    float4 acc_n = acc_s;

#pragma unroll 8
    for (int f = 0; f < NN_MAXF; ++f) {
        const float w  = vrow[f];                  // uniform -> SGPR
        const int   rs = srow[f];                  // uniform -> SGPR
        const int   rn = nrow[f];                  // uniform -> SGPR
        const float4 ws = *(const float4*)(ft_w + rs * NN_FTOUT + col);  // global_load_b128 (saddr+voffset)
        const float4 wn = *(const float4*)(ft_w + rn * NN_FTOUT + col);
        acc_s.x = fmaf(w, ws.x, acc_s.x);
        acc_s.y = fmaf(w, ws.y, acc_s.y);
        acc_s.z = fmaf(w, ws.z, acc_s.z);
        acc_s.w = fmaf(w, ws.w, acc_s.w);
        acc_n.x = fmaf(w, wn.x, acc_n.x);
        acc_n.y = fmaf(w, wn.y, acc_n.y);
        acc_n.z = fmaf(w, wn.z, acc_n.z);
        acc_n.w = fmaf(w, wn.w, acc_n.w);
    }

    // clipped ReLU
    acc_s.x = nn_clamp01(acc_s.x); acc_s.y = nn_clamp01(acc_s.y);
    acc_s.z = nn_clamp01(acc_s.z); acc_s.w = nn_clamp01(acc_s.w);
    acc_n.x = nn_clamp01(acc_n.x); acc_n.y = nn_clamp01(acc_n.y);
    acc_n.z = nn_clamp01(acc_n.z); acc_n.w = nn_clamp01(acc_n.w);

    // ---- Ensure staged weights are visible, then fused output layer ----
#ifdef NN_HAVE_WAIT_ASYNC
    __builtin_amdgcn_s_wait_asynccnt(0);  // own wave's async copies done
#endif
    __syncthreads();                      // all waves' LDS writes visible

    const float4 wo_s = *(const float4*)(s_outw + col);
    const float4 wo_n = *(const float4*)(s_outw + NN_FTOUT + col);

    float p = acc_s.x * wo_s.x + acc_s.y * wo_s.y +
              acc_s.z * wo_s.z + acc_s.w * wo_s.w +
              acc_n.x * wo_n.x + acc_n.y * wo_n.y +
              acc_n.z * wo_n.z + acc_n.w * wo_n.w;

    // wave32 reduction
#pragma unroll
    for (int off = 16; off >= 1; off >>= 1)
        p += __shfl_down(p, off, 32);

    const int wave = tid >> 5;
    const int lane = tid & 31;
    if (lane == 0) s_part[wave] = p;
    __syncthreads();

    if (tid == 0) {
        float t = 0.0f;
#pragma unroll
        for (int w = 0; w < 8; ++w) t += s_part[w];
        t += out_b[0];
        out[b] = 1.0f / (1.0f + __expf(-t));
    }
}

extern "C" void kernel_launch(void* const* d_in, const int* in_sizes, int n_in,
                              void* d_out, int out_size, void* d_ws, size_t ws_size,
                              hipStream_t stream) {
    (void)in_sizes; (void)n_in; (void)d_ws; (void)ws_size;
    const float* values   = (const float*)d_in[0];
    const int*   stm_idx  = (const int*)  d_in[1];
    const int*   nstm_idx = (const int*)  d_in[2];
    const float* ft_w     = (const float*)d_in[3];
    const float* ft_b     = (const float*)d_in[4];
    const float* out_w    = (const float*)d_in[5];
    const float* out_b    = (const float*)d_in[6];
    float*       out      = (float*)d_out;
    (void)out_size;

    nnue_fused_kernel<<<dim3(NN_B), dim3(256), 0, stream>>>(
        values, stm_idx, nstm_idx, ft_w, ft_b, out_w, out_b, out);
}